// GraphLayer_85650237817502
// MI455X (gfx1250) — compile-verified
//
#include <hip/hip_runtime.h>

typedef float v2f __attribute__((ext_vector_type(2)));
typedef float v8f __attribute__((ext_vector_type(8)));

#define NODE_F 64
#define EDGE_F 32
#define HIDDEN 64
#define EDGE_IN (NODE_F + EDGE_F)   // 96
#define NODE_IN (HIDDEN + NODE_F)   // 128

// ---------------------------------------------------------------------------
// Zero the scatter-add accumulator (harness poisons d_ws with 0xAA).
// ---------------------------------------------------------------------------
__global__ void zero_ws_kernel(float4* __restrict__ p, int n4) {
  int i = blockIdx.x * blockDim.x + threadIdx.x;
  if (i < n4) p[i] = float4{0.f, 0.f, 0.f, 0.f};
}

// ---------------------------------------------------------------------------
// Edge MLP: out_edge[e, :] = PReLU(concat(nf[sender[e]], ef[e]) @ W^T, alpha)
// plus scatter-add into node_acc[receiver[e], :].
// One wave per 16-edge tile; 4 x v8f accumulators cover all 64 outputs.
// fp32 WMMA 16x16x4: 24 K-steps x 4 N-tiles = 96 wmma per wave.
// ---------------------------------------------------------------------------
__global__ __launch_bounds__(256)
void edge_mlp_kernel(const float* __restrict__ nf,
                     const int*   __restrict__ ei,    // [2, E]: row0=recv, row1=send
                     const float* __restrict__ ef,    // [E, 32]
                     const float* __restrict__ W,     // [64, 96] row-major [out,in]
                     const float* __restrict__ alpha, // [64]
                     float* __restrict__ out_edge,    // [E, 64]
                     float* __restrict__ node_acc,    // [N, 64] (zeroed)
                     int E) {
  __shared__ float wlds[HIDDEN * EDGE_IN];            // 24 KB
  for (int i = threadIdx.x; i < HIDDEN * EDGE_IN; i += blockDim.x) wlds[i] = W[i];
  __syncthreads();

  const int lane = threadIdx.x & 31;
  const int wave = threadIdx.x >> 5;
  const int tile = blockIdx.x * 8 + wave;
  const int numTiles = (E + 15) >> 4;
  if (tile >= numTiles) return;                       // wave-uniform; EXEC stays full

  const int row = lane & 15;                          // A: M index / B: N index
  const int kh  = lane >> 4;                          // K-half selector (0/1)

  const int m_global  = tile * 16 + row;
  const int m_clamped = m_global < E ? m_global : E - 1;
  const int sender    = ei[E + m_clamped];            // senders = edge_index[1]

  const float* nrow = nf + (long)sender    * NODE_F;
  const float* erow = ef + (long)m_clamped * EDGE_F;

  v8f acc[4];
#pragma unroll
  for (int t = 0; t < 4; ++t)
#pragma unroll
    for (int j = 0; j < 8; ++j) acc[t][j] = 0.f;

#pragma unroll 4
  for (int k0 = 0; k0 < EDGE_IN; k0 += 4) {
    const int c = k0 + 2 * kh;                        // even; never straddles col 64
    v2f a = (c < NODE_F) ? *(const v2f*)(nrow + c)
                         : *(const v2f*)(erow + (c - NODE_F));
#pragma unroll
    for (int t = 0; t < 4; ++t) {
      const int n = t * 16 + row;
      v2f b = *(const v2f*)(&wlds[n * EDGE_IN + c]);  // B[k][n] = W[n][k]
      acc[t] = __builtin_amdgcn_wmma_f32_16x16x4_f32(
          false, a, false, b, (short)0, acc[t], false, false);
    }
  }

  // ------------------------------------------------------------------------
  // Epilogue. C/D layout: lane L holds col L&15; VGPR j holds row 8*(L>>4)+j.
  // Receivers for this lane-half are 8 consecutive ints -> two b128 loads.
  // ------------------------------------------------------------------------
  const int mBase = tile * 16 + kh * 8;               // first row for this half
  const int mSafe = mBase + 8 <= E ? mBase : (E >= 8 ? E - 8 : 0);
  const int4 r0 = *(const int4*)(ei + mSafe);
  const int4 r1 = *(const int4*)(ei + mSafe + 4);
  const int rcv[8] = {r0.x, r0.y, r0.z, r0.w, r1.x, r1.y, r1.z, r1.w};

  const bool fullTile = (tile * 16 + 16 <= E);        // wave-uniform
  if (fullTile) {
#pragma unroll
    for (int t = 0; t < 4; ++t) {
      const int n  = t * 16 + row;
      const float al = alpha[n];
#pragma unroll
      for (int j = 0; j < 8; ++j) {
        const float x = acc[t][j];
        const float y = x >= 0.f ? x : al * x;
        out_edge[(long)(mBase + j) * HIDDEN + n] = y;
        atomicAdd(&node_acc[(long)rcv[j] * HIDDEN + n], y);
      }
    }
  } else {                                            // tail tile (rare/never)
#pragma unroll
    for (int t = 0; t < 4; ++t) {
      const int n  = t * 16 + row;
      const float al = alpha[n];
#pragma unroll
      for (int j = 0; j < 8; ++j) {
        const int m = mBase + j;
        if (m < E) {
          const float x = acc[t][j];
          const float y = x >= 0.f ? x : al * x;
          out_edge[(long)m * HIDDEN + n] = y;
          atomicAdd(&node_acc[(long)ei[m] * HIDDEN + n], y);
        }
      }
    }
  }
}

// ---------------------------------------------------------------------------
// Node MLP: out_node[v, :] = PReLU(concat(node_acc[v], nf[v]) @ W^T, alpha)
// Same wave-per-16-row-tile structure; K = 128 -> 32 K-steps x 4 N-tiles.
// ---------------------------------------------------------------------------
__global__ __launch_bounds__(256)
void node_mlp_kernel(const float* __restrict__ node_acc, // [N, 64]
                     const float* __restrict__ nf,       // [N, 64]
                     const float* __restrict__ W,        // [64, 128]
                     const float* __restrict__ alpha,    // [64]
                     float* __restrict__ out_node,       // [N, 64]
                     int N) {
  __shared__ float wlds[HIDDEN * NODE_IN];               // 32 KB
  for (int i = threadIdx.x; i < HIDDEN * NODE_IN; i += blockDim.x) wlds[i] = W[i];
  __syncthreads();

  const int lane = threadIdx.x & 31;
  const int wave = threadIdx.x >> 5;
  const int tile = blockIdx.x * 8 + wave;
  const int numTiles = (N + 15) >> 4;
  if (tile >= numTiles) return;

  const int row = lane & 15;
  const int kh  = lane >> 4;

  const int v_global  = tile * 16 + row;
  const int v_clamped = v_global < N ? v_global : N - 1;

  const float* arow = node_acc + (long)v_clamped * HIDDEN;
  const float* frow = nf       + (long)v_clamped * NODE_F;

  v8f acc[4];
#pragma unroll
  for (int t = 0; t < 4; ++t)
#pragma unroll
    for (int j = 0; j < 8; ++j) acc[t][j] = 0.f;

#pragma unroll 4
  for (int k0 = 0; k0 < NODE_IN; k0 += 4) {
    const int c = k0 + 2 * kh;
    v2f a = (c < HIDDEN) ? *(const v2f*)(arow + c)
                         : *(const v2f*)(frow + (c - HIDDEN));
#pragma unroll
    for (int t = 0; t < 4; ++t) {
      const int n = t * 16 + row;
      v2f b = *(const v2f*)(&wlds[n * NODE_IN + c]);
      acc[t] = __builtin_amdgcn_wmma_f32_16x16x4_f32(
          false, a, false, b, (short)0, acc[t], false, false);
    }
  }

  const int mBase = tile * 16 + kh * 8;
  const bool fullTile = (tile * 16 + 16 <= N);          // wave-uniform
  if (fullTile) {
#pragma unroll
    for (int t = 0; t < 4; ++t) {
      const int n  = t * 16 + row;
      const float al = alpha[n];
#pragma unroll
      for (int j = 0; j < 8; ++j) {
        const float x = acc[t][j];
        out_node[(long)(mBase + j) * HIDDEN + n] = (x >= 0.f) ? x : al * x;
      }
    }
  } else {
#pragma unroll
    for (int t = 0; t < 4; ++t) {
      const int n  = t * 16 + row;
      const float al = alpha[n];
#pragma unroll
      for (int j = 0; j < 8; ++j) {
        const int m = mBase + j;
        if (m < N) {
          const float x = acc[t][j];
          out_node[(long)m * HIDDEN + n] = (x >= 0.f) ? x : al * x;
        }
      }
    }
  }
}

// ---------------------------------------------------------------------------
extern "C" void kernel_launch(void* const* d_in, const int* in_sizes, int n_in,
                              void* d_out, int out_size, void* d_ws, size_t ws_size,
                              hipStream_t stream) {
  const float* nf     = (const float*)d_in[0];
  const int*   ei     = (const int*)  d_in[1];
  const float* ef     = (const float*)d_in[2];
  const float* W_edge = (const float*)d_in[3];
  const float* a_edge = (const float*)d_in[4];
  const float* W_node = (const float*)d_in[5];
  const float* a_node = (const float*)d_in[6];

  const int N = in_sizes[0] / NODE_F;
  const int E = in_sizes[2] / EDGE_F;

  float* out_node = (float*)d_out;                       // [N, 64] first in tuple
  float* out_edge = (float*)d_out + (long)N * HIDDEN;    // [E, 64] second
  float* node_acc = (float*)d_ws;                        // [N, 64] scratch accumulator

  // 1) zero the segment-sum accumulator
  const int n4 = (N * HIDDEN) / 4;
  zero_ws_kernel<<<(n4 + 255) / 256, 256, 0, stream>>>((float4*)node_acc, n4);

  // 2) edge MLP + scatter-add (1 wave per 16 edges, 8 waves per block)
  const int eTiles = (E + 15) / 16;
  edge_mlp_kernel<<<(eTiles + 7) / 8, 256, 0, stream>>>(
      nf, ei, ef, W_edge, a_edge, out_edge, node_acc, E);

  // 3) node MLP
  const int nTiles = (N + 15) / 16;
  node_mlp_kernel<<<(nTiles + 7) / 8, 256, 0, stream>>>(
      node_acc, nf, W_node, a_node, out_node, N);
}